// CSCR_86011015070101
// MI455X (gfx1250) — compile-verified
//
#include <hip/hip_runtime.h>
#include <math.h>
#include <stdint.h>

// CSCR forward for MI455X (gfx1250, wave32).
// B=32, C=256, H=W=56. Memory-bound op: ~590MB min HBM traffic -> ~25us @ 23.3TB/s.
// Passes:
//   k0: zero the two atomicMax counters
//   k1: per-pixel channel mean/max -> sa_sig, plus deterministic per-block ||sa||^2 partials
//   k2: per-(stream,b,16ch) WMMA_F32_16X16X4_F32 GEMV vs sa_sig, x staged via
//       global_load_async_to_lds_b128 double-buffering (ASYNCcnt), normsq on same data
//   k3: per-(stream,b) bitonic argsort of sims (stable ties like jnp.argsort) + positive count
//   k5: channel gather (+ insert extra channel) * sa_sig, nontemporal stores
#define B_   32
#define C_   256
#define HW_  3136
#define CHW_ (C_ * HW_)
#define EPS_ 1e-12f
#define NB1  13           // ceil(HW_/256) blocks per batch in k1

typedef float v2f __attribute__((ext_vector_type(2)));
typedef float v8f __attribute__((ext_vector_type(8)));

// workspace layout in float units (total ~600KB)
#define OFF_SA    0        // sa_sig:  B*HW          = 100352
#define OFF_PART  100352   // ||sa||^2 partials: B*16 = 512
#define OFF_DOT   100864   // dot:     2*B*C         = 16384
#define OFF_NSQ   117248   // normsq:  2*B*C         = 16384
#define OFF_ORDER 133632   // order (int): 2*B*C     = 16384
#define OFF_NPOS  150016   // n_pos (int): 2

// ---------------------------------------------------------------- k0: init
__global__ void k0_init(int* __restrict__ npos) {
    if (threadIdx.x < 2) npos[threadIdx.x] = 0;
}

// ------------------------------------------------- k1: sa_sig + norm partials
// One thread per (b,hw); loops C channels with stride HW -> consecutive lanes
// hit consecutive addresses (fully coalesced). Reads 196MB.
__global__ void __launch_bounds__(256)
k1_sa(const float* __restrict__ rgb, const float* __restrict__ ir,
      float* __restrict__ sa_sig, float* __restrict__ part) {
    const int b  = blockIdx.y;
    const int hw = blockIdx.x * 256 + threadIdx.x;
    float ss = 0.f;
    if (hw < HW_) {
        const size_t base = (size_t)b * CHW_ + hw;
        float sr = 0.f, si = 0.f, mr = -INFINITY, mi = -INFINITY;
        for (int c = 0; c < C_; ++c) {
            float vr = rgb[base + (size_t)c * HW_];
            float vi = ir [base + (size_t)c * HW_];
            sr += vr; si += vi;
            mr = fmaxf(mr, vr); mi = fmaxf(mi, vi);
        }
        float sa = fmaxf(sr * (1.f / C_) + si * (1.f / C_), mr + mi);
        ss = 1.f / (1.f + expf(-sa));
        sa_sig[b * HW_ + hw] = ss;
    }
    __shared__ float red[256];
    red[threadIdx.x] = ss * ss;
    __syncthreads();
    for (int s = 128; s > 0; s >>= 1) {
        if (threadIdx.x < s) red[threadIdx.x] += red[threadIdx.x + s];
        __syncthreads();
    }
    if (threadIdx.x == 0) part[b * 16 + blockIdx.x] = red[0];  // deterministic (no atomics)
}

// ------------------------------------------------- k2: WMMA GEMV + normsq
#define TK 64                  // hw tile (3136 = 49*64)
#define NT (HW_ / TK)          // 49
#define XS (TK + 4)            // padded LDS row stride: (r*68+t)%64 distinct per row -> no bank conflicts

__device__ __forceinline__ void async_b128(const float* g, const float* l) {
    // GLOBAL_LOAD_ASYNC_TO_LDS_B128: per-lane 16B global->LDS, tracked by ASYNCcnt.
    unsigned loff = (unsigned)(uintptr_t)l;              // low 32 bits of flat addr = LDS offset
    unsigned long long ga = (unsigned long long)(uintptr_t)g;
    asm volatile("global_load_async_to_lds_b128 %0, %1, off"
                 :: "v"(loff), "v"(ga) : "memory");
}

__global__ void __launch_bounds__(32)
k2_dot(const float* __restrict__ rgb, const float* __restrict__ ir,
       const float* __restrict__ sa_sig,
       float* __restrict__ dot_o, float* __restrict__ nsq_o) {
    __shared__ __align__(16) float s_sa[HW_];            // full sa row for this batch (12.5KB)
    __shared__ __align__(16) float s_x[2 * 16 * XS];     // double-buffered 16ch x 64hw tiles

    const int c0 = blockIdx.x * 16, b = blockIdx.y, s = blockIdx.z;
    const float* xb  = (s ? ir : rgb) + (size_t)b * CHW_;
    const float* sab = sa_sig + b * HW_;
    const int lane = threadIdx.x;

    // Stage sa asynchronously (784 16B chunks, 25 wave issues).
    for (int i = lane; i < HW_ / 4; i += 32)
        async_b128(sab + i * 4, s_sa + i * 4);

    // Tile loader: 64 floats/row = 16 lanes x 16B -> lanes[0:15]=row 2rr, lanes[16:31]=row 2rr+1.
    const int sub = lane >> 4, col = (lane & 15) * 4;
    auto issue = [&](int t, int buf) {
        const int t0 = t * TK;
        for (int rr = 0; rr < 8; ++rr) {
            const int r = rr * 2 + sub;
            async_b128(xb + (size_t)(c0 + r) * HW_ + t0 + col,
                       s_x + buf * 16 * XS + r * XS + col);
        }
    };
    issue(0, 0);
    issue(1, 1);

    // WMMA F32 16x16x4: A[i][k] = x[c0+i, t+k], B[k][j] = sa[t+k] (broadcast over j).
    // Layouts per ISA: A vgpr0/1 = K{0,1} (lanes<16) / K{2,3} (lanes>=16); B assumed matching striping.
    const int ch = lane & 15, koff = (lane >> 4) * 2;
    v8f acc = {};
    float nsq = 0.f;
    for (int t = 0; t < NT; ++t) {
        if (t + 1 < NT) asm volatile("s_wait_asynccnt 0x8" ::: "memory");  // this tile done, next in flight
        else            asm volatile("s_wait_asynccnt 0x0" ::: "memory");
        const float* xs = s_x + (t & 1) * 16 * XS + ch * XS;
        const float* sp = s_sa + t * TK;
        #pragma unroll
        for (int k = 0; k < TK; k += 4) {
            const int k0 = k + koff;
            float x0 = xs[k0], x1 = xs[k0 + 1];
            v2f a;  a.x  = x0;      a.y  = x1;
            v2f bb; bb.x = sp[k0];  bb.y = sp[k0 + 1];
            nsq += x0 * x0 + x1 * x1;                    // normsq on the same staged data
            acc = __builtin_amdgcn_wmma_f32_16x16x4_f32(
                false, a, false, bb, (short)0, acc, false, false);
        }
        if (t + 2 < NT) {
            asm volatile("s_wait_dscnt 0x0" ::: "memory");  // buffer reads drained before overwrite
            issue(t + 2, t & 1);
        }
    }
    // normsq: lane c and lane c+16 hold complementary hw halves of channel c
    float tot = nsq + __shfl_xor(nsq, 16);
    const int obase = (s * B_ + b) * C_;
    if (lane < 16) nsq_o[obase + c0 + lane] = tot;
    // D layout: lane 0 holds (M=r,N=0) r=0..7; lane 16 holds (M=8+r,N=0). All columns equal.
    if (lane == 0) {
        for (int r = 0; r < 8; ++r) dot_o[obase + c0 + r] = acc[r];
    }
    if (lane == 16) {
        for (int r = 0; r < 8; ++r) dot_o[obase + c0 + 8 + r] = acc[r];
    }
}

// ------------------------------------------------- k3: sims + bitonic argsort
__global__ void __launch_bounds__(256)
k3_sort(const float* __restrict__ dot_i, const float* __restrict__ nsq_i,
        const float* __restrict__ part,
        int* __restrict__ order, int* __restrict__ npos) {
    const int b = blockIdx.x, s = blockIdx.y, tid = threadIdx.x;
    __shared__ float kk[256];
    __shared__ int   vv[256];
    __shared__ int   ired[256];
    __shared__ float nsa_s;
    if (tid == 0) {
        float acc = 0.f;
        for (int i = 0; i < NB1; ++i) acc += part[b * 16 + i];   // deterministic order
        nsa_s = fmaxf(sqrtf(acc), EPS_);
    }
    __syncthreads();
    const int base = (s * B_ + b) * C_;
    float nx  = fmaxf(sqrtf(nsq_i[base + tid]), EPS_);
    float sim = dot_i[base + tid] / (nsa_s * nx);
    kk[tid] = sim; vv[tid] = tid;
    __syncthreads();
    // ascending bitonic sort on (sim, idx) lexicographic -> stable like jnp.argsort
    for (int len = 2; len <= 256; len <<= 1)
        for (int j = len >> 1; j > 0; j >>= 1) {
            const int ixj = tid ^ j;
            if (ixj > tid) {
                const bool up = ((tid & len) == 0);
                float ka = kk[tid], kb = kk[ixj];
                int   va = vv[tid], vb = vv[ixj];
                bool agtb = (ka > kb) || (ka == kb && va > vb);
                if (up == agtb) { kk[tid] = kb; kk[ixj] = ka; vv[tid] = vb; vv[ixj] = va; }
            }
            __syncthreads();
        }
    order[base + tid] = vv[tid];
    ired[tid] = (kk[tid] > 0.f) ? 1 : 0;
    __syncthreads();
    for (int st = 128; st > 0; st >>= 1) {
        if (tid < st) ired[tid] += ired[tid + st];
        __syncthreads();
    }
    if (tid == 0) atomicMax(&npos[s], ired[0]);   // int max: order-independent, deterministic
}

// ------------------------------------------------- k5: gather + gate, NT stores
__global__ void __launch_bounds__(256)
k5_out(const float* __restrict__ rgb, const float* __restrict__ ir,
       const float* __restrict__ sa_sig, const int* __restrict__ order,
       const int* __restrict__ npos, float* __restrict__ out) {
    const int gid = blockIdx.x;
    const int s   = gid / (B_ * C_);
    const int rem = gid - s * B_ * C_;
    const int b   = rem / C_;
    const int c   = rem - b * C_;
    const int n_self = npos[s], n_other = npos[1 - s];
    const int obase  = (s * B_ + b) * C_;
    bool extra = false;
    int  srcC  = 0;
    if (n_other > n_self) {                 // insert extra channel at n_self, shift, drop last
        if (c < n_self)       srcC = order[obase + c];
        else if (c == n_self) extra = true;
        else                  srcC = order[obase + c - 1];
    } else {
        srcC = order[obase + c];
    }
    const float* x  = s ? ir : rgb;
    const float* xr = rgb + ((size_t)b * C_ + order[b * C_]) * HW_;             // rgb sorted[0]
    const float* xi = ir  + ((size_t)b * C_ + order[(B_ + b) * C_]) * HW_;      // ir  sorted[0]
    const float* xc = x   + ((size_t)b * C_ + srcC) * HW_;
    const float* sab = sa_sig + b * HW_;
    float* o = out + (size_t)gid * HW_;
    for (int hw = threadIdx.x; hw < HW_; hw += 256) {
        float sa = sab[hw];
        float v  = extra ? fmaxf(xr[hw], xi[hw]) : xc[hw];
        __builtin_nontemporal_store(v * sa, o + hw);   // streamed output, no reuse
    }
}

// ---------------------------------------------------------------- launch
extern "C" void kernel_launch(void* const* d_in, const int* in_sizes, int n_in,
                              void* d_out, int out_size, void* d_ws, size_t ws_size,
                              hipStream_t stream) {
    const float* rgb = (const float*)d_in[0];
    const float* ir  = (const float*)d_in[1];
    float* ws    = (float*)d_ws;
    float* sa    = ws + OFF_SA;
    float* part  = ws + OFF_PART;
    float* dotb  = ws + OFF_DOT;
    float* nsqb  = ws + OFF_NSQ;
    int*   order = (int*)(ws + OFF_ORDER);
    int*   npos  = (int*)(ws + OFF_NPOS);
    float* out   = (float*)d_out;

    k0_init<<<1, 64, 0, stream>>>(npos);
    k1_sa  <<<dim3(NB1, B_),        256, 0, stream>>>(rgb, ir, sa, part);
    k2_dot <<<dim3(C_ / 16, B_, 2),  32, 0, stream>>>(rgb, ir, sa, dotb, nsqb);
    k3_sort<<<dim3(B_, 2),          256, 0, stream>>>(dotb, nsqb, part, order, npos);
    k5_out <<<2 * B_ * C_,          256, 0, stream>>>(rgb, ir, sa, order, npos, out);
}